// GlobalAttentionPoolingHead_64398739636495
// MI455X (gfx1250) — compile-verified
//
#include <hip/hip_runtime.h>
#include <math.h>

#define B_  32
#define S_  2048
#define D_  1024
#define EPSN 1e-12f

typedef float v2f __attribute__((ext_vector_type(2)));
typedef float v8f __attribute__((ext_vector_type(8)));
typedef int   v4i __attribute__((ext_vector_type(4)));

// ---------------- CDNA5 async global->LDS copy (ASYNCcnt-tracked) ----------
#if __has_builtin(__builtin_amdgcn_global_load_async_to_lds_b128)
#define HAVE_ASYNC_LDS 1
#else
#define HAVE_ASYNC_LDS 0
#endif

__device__ __forceinline__ void cp_g2l_16B(const float* __restrict__ g, float* l) {
#if HAVE_ASYNC_LDS
    __builtin_amdgcn_global_load_async_to_lds_b128(
        (__attribute__((address_space(1))) v4i*)g,
        (__attribute__((address_space(3))) v4i*)l,
        0, 0);
#else
    *(float4*)l = *(const float4*)g;
#endif
}

__device__ __forceinline__ void async_wait_all() {
#if HAVE_ASYNC_LDS
#if __has_builtin(__builtin_amdgcn_s_wait_asynccnt)
    __builtin_amdgcn_s_wait_asynccnt(0);
#else
    asm volatile("s_wait_asynccnt 0x0" ::: "memory");
#endif
#endif
}

// ------------------------- main fused GEMM + attention reductions (K4)
// grid (D/128, S/64, B); block 256 = 8 waves. Double-buffered LDS stages,
// async global->LDS copies (TA unit) overlapped with WMMA issue.
// No ragged mask needed: ssq/dot at s>=len are never read downstream.
#define BM 64
#define BN 128
#define KC 32
#define AP 36   // A pitch: row byte pitch 144 (16B aligned for async B128),
                // 36*lr mod 64 spreads 16 lanes over 16 banks, float2 stays 8B aligned

__global__ __launch_bounds__(256)
void k_local_attn(const float* __restrict__ X,
                  const float* __restrict__ Wl, const float* __restrict__ gp,
                  float* __restrict__ ssq, float* __restrict__ dotv) {
    __shared__ float lA[2][BM * AP];
    __shared__ float lB[2][KC * BN];
    int b    = blockIdx.z;
    int s0   = blockIdx.y * BM;
    int nblk = blockIdx.x * BN;
    int tid  = threadIdx.x;
    int wave = tid >> 5, lane = tid & 31, lr = lane & 15, half = lane >> 4;
    int m0w = (wave >> 1) * 16;
    int n0w = (wave & 1) * 64;
    v8f acc[4] = {v8f{}, v8f{}, v8f{}, v8f{}};

    const float* Xb = X + ((size_t)b * S_ + s0) * D_;
    const float* Wb = Wl + nblk;
    int ar = tid >> 3, ac4 = (tid & 7) * 4;    // A: 64 rows x 32 k, 2 x 16B/thread
    int br = tid >> 5, bc4 = (tid & 31) * 4;   // B: 32 rows x 128 n, 4 x 16B/thread

    auto issue_stage = [&](int k0, int stg) {
#pragma unroll
        for (int rep = 0; rep < 2; ++rep) {
            int row = ar + rep * 32;
            cp_g2l_16B(Xb + (size_t)row * D_ + k0 + ac4, &lA[stg][row * AP + ac4]);
        }
#pragma unroll
        for (int rep = 0; rep < 4; ++rep) {
            int row = br + rep * 8;
            cp_g2l_16B(Wb + (size_t)(k0 + row) * D_ + bc4, &lB[stg][row * BN + bc4]);
        }
    };

    issue_stage(0, 0);
    int buf = 0;
    for (int k0 = 0; k0 < D_; k0 += KC) {
        async_wait_all();          // this wave's stage-k0 copies have landed
        __syncthreads();           // every wave's copies have landed
        if (k0 + KC < D_) issue_stage(k0 + KC, buf ^ 1);
        if (k0 + 2 * KC < D_)      // warm L2 for the stage after next (X stream)
            __builtin_prefetch(Xb + (size_t)ar * D_ + k0 + 2 * KC, 0, 1);
        const float* As = lA[buf];
        const float* Bs = lB[buf];
#pragma unroll
        for (int k = 0; k < KC; k += 4) {
            int kk = k + 2 * half;
            v2f a = *(const v2f*)&As[(m0w + lr) * AP + kk];
#pragma unroll
            for (int t = 0; t < 4; ++t) {
                v2f bb;
                bb.x = Bs[kk * BN + n0w + 16 * t + lr];
                bb.y = Bs[(kk + 1) * BN + n0w + 16 * t + lr];
                acc[t] = __builtin_amdgcn_wmma_f32_16x16x4_f32(
                    false, a, false, bb, (short)0, acc[t], false, false);
            }
        }
        buf ^= 1;
    }

    // Epilogue: per-token partial ||y||^2 and y.gp over this 128-col slice.
    float gpv[4];
#pragma unroll
    for (int t = 0; t < 4; ++t)
        gpv[t] = gp[b * D_ + nblk + n0w + 16 * t + lr];
#pragma unroll
    for (int i = 0; i < 8; ++i) {
        float ps = 0.f, pd = 0.f;
#pragma unroll
        for (int t = 0; t < 4; ++t) {
            float y = acc[t][i];
            ps += y * y;
            pd += y * gpv[t];
        }
#pragma unroll
        for (int off = 1; off < 16; off <<= 1) {
            ps += __shfl_xor(ps, off, 32);
            pd += __shfl_xor(pd, off, 32);
        }
        if (lr == 0) {
            int s = s0 + m0w + i + 8 * half;
            atomicAdd(&ssq[b * S_ + s], ps);
            atomicAdd(&dotv[b * S_ + s], pd);
        }
    }
}

// ---------------------------------------------------------------- zero fill
__global__ void k_zero(float* __restrict__ p, int n) {
    int i = blockIdx.x * blockDim.x + threadIdx.x;
    if (i < n) p[i] = 0.0f;
}

// ------------------------------------------------- masked sequence sum (K1)
__global__ void k_seqsum(const float* __restrict__ X, const int* __restrict__ len,
                         float* __restrict__ seqsum) {
    int b = blockIdx.x, sc = blockIdx.y;
    int L = len[b];
    int s0 = sc * 256;
    int rem = L - s0;
    if (rem <= 0) return;
    int n = rem < 256 ? rem : 256;
    int t = threadIdx.x;
    const float4* Xp = (const float4*)(X + ((size_t)b * S_ + s0) * D_);
    float4 acc = {0.f, 0.f, 0.f, 0.f};
    for (int j = 0; j < n; ++j) {
        float4 v = Xp[(size_t)j * (D_ / 4) + t];
        acc.x += v.x; acc.y += v.y; acc.z += v.z; acc.w += v.w;
    }
    float* o = seqsum + b * D_ + t * 4;
    atomicAdd(o + 0, acc.x); atomicAdd(o + 1, acc.y);
    atomicAdd(o + 2, acc.z); atomicAdd(o + 3, acc.w);
}

// ------------------------------------- g = seqsum @ W_global via WMMA (K2)
__global__ void k_global_proj(const float* __restrict__ A,   // [32,1024]
                              const float* __restrict__ Wg,  // [1024,1024]
                              float* __restrict__ g) {       // [32,1024]
    int tid = threadIdx.x;
    int w = tid >> 5, lane = tid & 31, lr = lane & 15, half = lane >> 4;
    int m0  = blockIdx.y * 16;
    int n0w = blockIdx.x * 64 + w * 32;
    v8f acc0 = {}; v8f acc1 = {};
    for (int k = 0; k < D_; k += 4) {
        int kk = k + 2 * half;
        v2f a;  a.x  = A[(m0 + lr) * D_ + kk];
                a.y  = A[(m0 + lr) * D_ + kk + 1];
        v2f b0; b0.x = Wg[(size_t)kk * D_ + n0w + lr];
                b0.y = Wg[(size_t)(kk + 1) * D_ + n0w + lr];
        v2f b1; b1.x = Wg[(size_t)kk * D_ + n0w + 16 + lr];
                b1.y = Wg[(size_t)(kk + 1) * D_ + n0w + 16 + lr];
        acc0 = __builtin_amdgcn_wmma_f32_16x16x4_f32(false, a, false, b0, (short)0, acc0, false, false);
        acc1 = __builtin_amdgcn_wmma_f32_16x16x4_f32(false, a, false, b1, (short)0, acc1, false, false);
    }
#pragma unroll
    for (int i = 0; i < 8; ++i) {
        int row = m0 + i + 8 * half;
        g[row * D_ + n0w + lr]      = acc0[i];
        g[row * D_ + n0w + 16 + lr] = acc1[i];
    }
}

// --------------------------------------------- per-row l2 normalize (K3)
__global__ void k_l2norm_rows(float* __restrict__ g) {
    __shared__ float red[256];
    int b = blockIdx.x, t = threadIdx.x;
    float4 v = ((const float4*)(g + b * D_))[t];
    red[t] = v.x * v.x + v.y * v.y + v.z * v.z + v.w * v.w;
    __syncthreads();
    for (int o = 128; o > 0; o >>= 1) {
        if (t < o) red[t] += red[t + o];
        __syncthreads();
    }
    float scale = rsqrtf(fmaxf(red[0], EPSN));
    float4 o4 = {v.x * scale, v.y * scale, v.z * scale, v.w * scale};
    ((float4*)(g + b * D_))[t] = o4;
}

// ----------------------------- attention-weighted ragged sum (K5)
__global__ void k_weighted_sum(const float* __restrict__ X, const int* __restrict__ len,
                               const float* __restrict__ ssq, const float* __restrict__ dotv,
                               float* __restrict__ out) {
    __shared__ float att[256];
    int b = blockIdx.x, sc = blockIdx.y;
    int L = len[b];
    int s0 = sc * 256;
    int rem = L - s0;
    if (rem <= 0) return;
    int n = rem < 256 ? rem : 256;
    int t = threadIdx.x;
    {
        float a = 0.f;
        if (t < n) {
            int s = s0 + t;
            a = dotv[b * S_ + s] * rsqrtf(fmaxf(ssq[b * S_ + s], EPSN));
        }
        att[t] = a;
    }
    __syncthreads();
    const float4* Xp = (const float4*)(X + ((size_t)b * S_ + s0) * D_);
    float4 acc = {0.f, 0.f, 0.f, 0.f};
    for (int j = 0; j < n; ++j) {
        float w = att[j];
        float4 v = Xp[(size_t)j * (D_ / 4) + t];
        acc.x += w * v.x; acc.y += w * v.y;
        acc.z += w * v.z; acc.w += w * v.w;
    }
    float* o = out + b * D_ + t * 4;
    atomicAdd(o + 0, acc.x); atomicAdd(o + 1, acc.y);
    atomicAdd(o + 2, acc.z); atomicAdd(o + 3, acc.w);
}

// ----------------------------------------------------------------- launch
extern "C" void kernel_launch(void* const* d_in, const int* in_sizes, int n_in,
                              void* d_out, int out_size, void* d_ws, size_t ws_size,
                              hipStream_t stream) {
    (void)in_sizes; (void)n_in; (void)out_size; (void)ws_size;
    const float* X       = (const float*)d_in[0];   // [32,2048,1024]
    const int*   lengths = (const int*)d_in[1];     // [32]
    const float* Wg      = (const float*)d_in[2];   // [1024,1024]
    const float* Wl      = (const float*)d_in[3];   // [1024,1024]
    float* out = (float*)d_out;                     // [32,1024]

    char* ws = (char*)d_ws;
    float* seqsum = (float*)(ws + 0);        // 32*1024 f32
    float* g      = (float*)(ws + 131072);   // 32*1024 f32 (g -> gp in place)
    float* ssq    = (float*)(ws + 262144);   // 32*2048 f32
    float* dotv   = (float*)(ws + 524288);   // 32*2048 f32

    k_zero<<<dim3(768), 256, 0, stream>>>((float*)ws, 196608);
    k_zero<<<dim3(128), 256, 0, stream>>>(out, B_ * D_);

    k_seqsum<<<dim3(B_, 8), 256, 0, stream>>>(X, lengths, seqsum);
    k_global_proj<<<dim3(16, 2), 64, 0, stream>>>(seqsum, Wg, g);
    k_l2norm_rows<<<dim3(B_), 256, 0, stream>>>(g);
    k_local_attn<<<dim3(D_ / BN, S_ / BM, B_), 256, 0, stream>>>(X, Wl, g, ssq, dotv);
    k_weighted_sum<<<dim3(B_, 8), 256, 0, stream>>>(X, lengths, ssq, dotv, out);
}